// RGCN_28819230556557
// MI455X (gfx1250) — compile-verified
//
#include <hip/hip_runtime.h>

typedef __attribute__((ext_vector_type(2))) float v2f;
typedef __attribute__((ext_vector_type(8))) float v8f;

#define NB    40   // N_BASES
#define EMBD  16   // EMB == N_CLASSES == 16 (WMMA tile width)
#define JPW   8    // j-tiles per wave in the WMMA GEMM (A-register reuse)

// ---------------------------------------------------------------------------
// Init: zero degree counts, zero h, seed out with bias2 (atomics add on top).
// ---------------------------------------------------------------------------
__global__ void init_kernel(unsigned* __restrict__ cnt, int cntN,
                            float* __restrict__ h, float* __restrict__ out,
                            const float* __restrict__ bias2, int hN) {
    int i = blockIdx.x * blockDim.x + threadIdx.x;
    if (i < cntN) cnt[i] = 0u;
    if (i < hN) {
        h[i]   = 0.0f;
        out[i] = bias2[i & 15];
    }
}

// ---------------------------------------------------------------------------
// Degree counts of the enriched graph (self-loops always have count 1,
// so only the 2E original+inverse edges are counted).
// ---------------------------------------------------------------------------
__global__ void count_kernel(const int* __restrict__ src, const int* __restrict__ dst,
                             const int* __restrict__ rel, unsigned* __restrict__ cnt,
                             int E, int N, int R) {
    int i = blockIdx.x * blockDim.x + threadIdx.x;
    if (i >= 2 * E) return;
    int s, r;
    if (i < E) { s = src[i];     r = rel[i];         }
    else       { s = dst[i - E]; r = rel[i - E] + R; }
    atomicAdd(&cnt[(size_t)r * N + s], 1u);
}

// ---------------------------------------------------------------------------
// w1[r, j] = sum_k comps1[r, k] * bases1[k, j]   (j flattens (node, emb))
// GEMM  [R_tot x 40] x [40 x JN]  via  V_WMMA_F32_16X16X4_F32.
// One wave computes JPW 16x16 output tiles: rows = relations (padded to 48),
// cols = 16 consecutive flattened columns. K = 40 -> 10 WMMA steps/tile.
//
// A is preloaded branch-free into 10 float2 registers (row clamped to a valid
// address, masked to zero for padding rows) and reused across all JPW tiles.
//
// Layouts (CDNA5 ISA 7.12.2, 32-bit):
//   A 16x4 : lane L holds A[L%16][(L/16)*2 + v]      (v = vector elem 0..1)
//   B 4x16 : lane L holds B[(L/16)*2 + v][L%16]
//   D 16x16: lane L, elem v holds D[v + (L/16)*8][L%16]
// ---------------------------------------------------------------------------
__global__ void w1_wmma_kernel(const float* __restrict__ comps1,
                               const float* __restrict__ bases1,
                               float* __restrict__ w1,
                               int R_tot, int JN, int MT, int JTO) {
    int lane = threadIdx.x & 31;
    int wave = (blockIdx.x * blockDim.x + threadIdx.x) >> 5;
    if (wave >= MT * JTO) return;
    int m     = wave / JTO;               // relation tile
    int jt0   = (wave % JTO) * JPW;       // first column tile for this wave
    int col   = lane & 15;
    int khalf = lane >> 4;                // 0 or 1
    int arow  = m * 16 + col;             // comps1 row this lane supplies for A

    // Branch-free A preload: clamp row to valid address, zero via mask.
    int   crow  = (arow < R_tot) ? arow : (R_tot - 1);
    float amask = (arow < R_tot) ? 1.0f : 0.0f;
    v2f a[NB / 4];
#pragma unroll
    for (int kk = 0; kk < NB / 4; ++kk) {
        int k0 = kk * 4 + khalf * 2;
        a[kk].x = amask * comps1[crow * NB + k0];
        a[kk].y = amask * comps1[crow * NB + k0 + 1];
    }

    for (int t = 0; t < JPW; ++t) {
        int jt = jt0 + t;
        if (jt * 16 >= JN) break;
        int j = jt * 16 + col;

        v8f acc = {};
#pragma unroll
        for (int kk = 0; kk < NB / 4; ++kk) {
            int k0 = kk * 4 + khalf * 2;
            v2f b;
            b.x = bases1[(size_t)k0 * JN + j];
            b.y = bases1[(size_t)(k0 + 1) * JN + j];
            acc = __builtin_amdgcn_wmma_f32_16x16x4_f32(
                /*neg_a=*/false, a[kk], /*neg_b=*/false, b,
                /*c_mod=*/(short)0, acc, /*reuse_a=*/false, /*reuse_b=*/false);
        }
#pragma unroll
        for (int vv = 0; vv < 8; ++vv) {
            int r = m * 16 + khalf * 8 + vv;
            if (r < R_tot) w1[(size_t)r * JN + j] = acc[vv];
        }
    }
}

// ---------------------------------------------------------------------------
// w2[r, e, c] = sum_b comps2[r, b] * bases2[b, e, c]   (tiny: 41*256 outputs)
// ---------------------------------------------------------------------------
__global__ void w2_kernel(const float* __restrict__ comps2,
                          const float* __restrict__ bases2,
                          float* __restrict__ w2, int R_tot) {
    int idx = blockIdx.x * blockDim.x + threadIdx.x;
    if (idx >= R_tot * 256) return;
    int r  = idx >> 8;
    int ec = idx & 255;
    float acc = 0.0f;
#pragma unroll
    for (int b = 0; b < NB; ++b)
        acc = fmaf(comps2[r * NB + b], bases2[b * 256 + ec], acc);
    w2[idx] = acc;
}

// ---------------------------------------------------------------------------
// Enriched-edge decode shared by both scatter layers.
// ---------------------------------------------------------------------------
__device__ __forceinline__ void decode_edge(int edge, int E, int N, int R,
                                            const int* __restrict__ src,
                                            const int* __restrict__ dst,
                                            const int* __restrict__ rel,
                                            int& s, int& o, int& r, bool& isLoop) {
    if (edge < E)            { s = src[edge];     o = dst[edge];     r = rel[edge];         isLoop = false; }
    else if (edge < 2 * E)   { s = dst[edge - E]; o = src[edge - E]; r = rel[edge - E] + R; isLoop = false; }
    else                     { s = edge - 2 * E;  o = s;             r = 2 * R;             isLoop = true;  }
}

// ---------------------------------------------------------------------------
// Layer 1:  h[s, c] += v * w1[r, o, c]     (16 lanes per edge, lane = channel)
// w1 (131 MB) is L2-resident on MI455X (192 MB L2), so gathers stay on-chip.
// ---------------------------------------------------------------------------
__global__ void layer1_kernel(const int* __restrict__ src, const int* __restrict__ dst,
                              const int* __restrict__ rel, const unsigned* __restrict__ cnt,
                              const float* __restrict__ w1, float* __restrict__ h,
                              int E, int N, int R) {
    int gid  = blockIdx.x * blockDim.x + threadIdx.x;
    int edge = gid >> 4;
    int c    = gid & 15;
    if (edge >= 2 * E + N) return;
    int s, o, r; bool isLoop;
    decode_edge(edge, E, N, R, src, dst, rel, s, o, r, isLoop);
    float v = isLoop ? 1.0f : 1.0f / (float)cnt[(size_t)r * N + s];
    float wv = w1[((size_t)r * N + o) * EMBD + c];
    atomicAdd(&h[s * EMBD + c], v * wv);
}

__global__ void relu_kernel(float* __restrict__ h, const float* __restrict__ bias1, int n) {
    int i = blockIdx.x * blockDim.x + threadIdx.x;
    if (i >= n) return;
    h[i] = fmaxf(h[i] + bias1[i & 15], 0.0f);
}

// ---------------------------------------------------------------------------
// Layer 2 (hv never materialized):
//   out[s, c] += v * sum_e h[o, e] * w2[r, e, c]
// 16 lanes per edge; lane c loads h[o, c], the 16x16 matvec uses cross-lane
// shuffles; w2 (41*256 floats = 42 KB) lives in LDS (320 KB/WGP).
// ---------------------------------------------------------------------------
__global__ void layer2_kernel(const int* __restrict__ src, const int* __restrict__ dst,
                              const int* __restrict__ rel, const unsigned* __restrict__ cnt,
                              const float* __restrict__ h, const float* __restrict__ w2g,
                              float* __restrict__ out, int E, int N, int R, int R_tot) {
    extern __shared__ float w2s[];            // R_tot * 256
    for (int i = threadIdx.x; i < R_tot * 256; i += blockDim.x) w2s[i] = w2g[i];
    __syncthreads();

    int gid  = blockIdx.x * blockDim.x + threadIdx.x;
    int edge = gid >> 4;
    int c    = gid & 15;
    if (edge >= 2 * E + N) return;
    int s, o, r; bool isLoop;
    decode_edge(edge, E, N, R, src, dst, rel, s, o, r, isLoop);
    float v  = isLoop ? 1.0f : 1.0f / (float)cnt[(size_t)r * N + s];
    float hc = h[o * EMBD + c];

    float acc = 0.0f;
    const float* w2r = &w2s[r * 256];
#pragma unroll
    for (int e = 0; e < 16; ++e) {
        float he = __shfl(hc, e, 16);         // h[o, e] from lane e of this 16-group
        acc = fmaf(he, w2r[e * 16 + c], acc);
    }
    atomicAdd(&out[s * EMBD + c], v * acc);
}

// ---------------------------------------------------------------------------
extern "C" void kernel_launch(void* const* d_in, const int* in_sizes, int n_in,
                              void* d_out, int out_size, void* d_ws, size_t ws_size,
                              hipStream_t stream) {
    const int*   src    = (const int*)d_in[0];
    const int*   dst    = (const int*)d_in[1];
    const int*   rel    = (const int*)d_in[2];
    const float* comps1 = (const float*)d_in[3];
    const float* bases1 = (const float*)d_in[4];
    const float* comps2 = (const float*)d_in[5];
    const float* bases2 = (const float*)d_in[6];
    const float* bias1  = (const float*)d_in[7];
    const float* bias2  = (const float*)d_in[8];
    float*       out    = (float*)d_out;

    const int E     = in_sizes[0];
    const int R_tot = in_sizes[3] / NB;              // comps1 [R_tot, 40]
    const int R     = (R_tot - 1) / 2;
    const int N     = in_sizes[4] / (NB * EMBD);     // bases1 [40, N, 16]
    const int JN    = N * EMBD;                      // flattened GEMM columns

    // Workspace layout
    char*  ws  = (char*)d_ws;
    size_t off = 0;
    float*    w1  = (float*)(ws + off); off += (size_t)R_tot * JN * sizeof(float);
    off = (off + 255) & ~(size_t)255;
    unsigned* cnt = (unsigned*)(ws + off); off += (size_t)2 * R * N * sizeof(unsigned);
    off = (off + 255) & ~(size_t)255;
    float*    h   = (float*)(ws + off); off += (size_t)N * EMBD * sizeof(float);
    off = (off + 255) & ~(size_t)255;
    float*    w2  = (float*)(ws + off);

    const int BLK = 256;

    // 1) init cnt / h / out
    {
        int cntN = 2 * R * N, hN = N * EMBD;
        int n = cntN > hN ? cntN : hN;
        init_kernel<<<(n + BLK - 1) / BLK, BLK, 0, stream>>>(cnt, cntN, h, out, bias2, hN);
    }
    // 2) degree counts
    count_kernel<<<(2 * E + BLK - 1) / BLK, BLK, 0, stream>>>(src, dst, rel, cnt, E, N, R);
    // 3) w1 GEMM via WMMA
    {
        int MT  = (R_tot + 15) / 16;
        int JT  = JN / 16;
        int JTO = (JT + JPW - 1) / JPW;              // outer column-tile groups
        long long threads = (long long)MT * JTO * 32;
        w1_wmma_kernel<<<(int)((threads + BLK - 1) / BLK), BLK, 0, stream>>>(
            comps1, bases1, w1, R_tot, JN, MT, JTO);
    }
    // 4) w2 (tiny)
    w2_kernel<<<(R_tot * 256 + BLK - 1) / BLK, BLK, 0, stream>>>(comps2, bases2, w2, R_tot);
    // 5) layer 1 scatter
    {
        long long threads = (long long)(2 * E + N) * 16;
        layer1_kernel<<<(int)((threads + BLK - 1) / BLK), BLK, 0, stream>>>(
            src, dst, rel, cnt, w1, h, E, N, R);
    }
    // 6) bias + relu
    relu_kernel<<<(N * EMBD + BLK - 1) / BLK, BLK, 0, stream>>>(h, bias1, N * EMBD);
    // 7) layer 2 scatter (out pre-seeded with bias2)
    {
        long long threads = (long long)(2 * E + N) * 16;
        size_t shbytes = (size_t)R_tot * 256 * sizeof(float);
        layer2_kernel<<<(int)((threads + BLK - 1) / BLK), BLK, shbytes, stream>>>(
            src, dst, rel, cnt, h, w2, out, E, N, R, R_tot);
    }
}